// Multi_RNNAttModel_33036888441447
// MI455X (gfx1250) — compile-verified
//
#include <hip/hip_runtime.h>
#include <hip/hip_bf16.h>

// ---------------- model constants ----------------
#define VSZ 2070
#define DSZ 512
#define SMAX 8
#define NMAX 50
#define ASZ 49
#define BSZ 64
#define TT 49              // NMAX-1 decoded steps
#define MROWS (SMAX*BSZ)   // 512 fused decode rows (s*64+b)

typedef __attribute__((ext_vector_type(2))) float v2f;
typedef __attribute__((ext_vector_type(8))) float v8f;

__device__ __forceinline__ float sigmoidf_(float x) { return 1.0f / (1.0f + __expf(-x)); }

__device__ __forceinline__ v8f wmma4(v2f a, v2f b, v8f c) {
    return __builtin_amdgcn_wmma_f32_16x16x4_f32(
        /*neg_a=*/false, a, /*neg_b=*/false, b,
        /*c_mod=*/(short)0, c, /*reuse_a=*/false, /*reuse_b=*/false);
}

// epilogue store of one 16x16 accumulator tile
__device__ __forceinline__ void store_tile(v8f acc, int m_base, int half, int nIdx,
                                           bool nOK, float bv, float* __restrict__ C,
                                           long long ldc, int act, int row_remap)
{
    if (!nOK) return;
#pragma unroll
    for (int r = 0; r < 8; ++r) {
        float v = acc[r] + bv;
        if (act == 1)      v = fmaxf(v, 0.f);
        else if (act == 2) v = tanhf(v);
        const int mr = m_base + r + (half << 3);   // VGPR r -> rows m_base+r / m_base+8+r
        long long orow = row_remap ? (long long)((mr & 63) * SMAX + (mr >> 6))
                                   : (long long)mr;
        C[orow * ldc + nIdx] = v;
    }
}

// ---------------------------------------------------------------------------
// Register-tiled WMMA-f32 GEMM:  C[M,N] = act( A[M,K] @ Bm[N,K]^T + bias[N] )
//   One wave computes a 32x32 output supertile as 2x2 WMMA 16x16 tiles:
//   per K-step (4): 2 A-frag + 2 B-frag loads feed 4 v_wmma_f32_16x16x4_f32.
//   Requirements (hold at every call site): M % 32 == 0, K % 4 == 0.
//   N edge (2070) handled by per-subtile clamp + predicated stores.
//   row_remap==1: output row = (m&63)*8 + (m>>6)  (s*64+b -> b*8+s for logits)
//   act: 0 none, 1 relu, 2 tanh
// ---------------------------------------------------------------------------
__global__ void gemm_wmma_f32(const float* __restrict__ A, int lda,
                              const float* __restrict__ Bm, int ldb,
                              const float* __restrict__ bias,
                              float* __restrict__ C, long long ldc,
                              int M, int N, int K, int act, int row_remap)
{
    const int wave = threadIdx.x >> 5;
    const int lane = threadIdx.x & 31;
    const int tiles_n = (N + 31) >> 5;
    const int tiles_m = M >> 5;
    const int tile = blockIdx.x * (blockDim.x >> 5) + wave;
    if (tile >= tiles_m * tiles_n) return;

    const int tm = tile / tiles_n;
    const int tn = tile - tm * tiles_n;
    const int m0 = tm << 5;
    const int n0 = tn << 5;

    const int half = lane >> 4;     // K sub-pair selector (K+0,1 vs K+2,3)
    const int l16  = lane & 15;

    // A fragments: rows m0+l16 and m0+16+l16
    const float* __restrict__ A0 = A + (long long)(m0 + l16) * lda + (half << 1);
    const float* __restrict__ A1 = A0 + (long long)16 * lda;
    // B fragments: columns n0+l16 and n0+16+l16 (rows of Bm)
    const int  n0i = n0 + l16;
    const int  n1i = n0 + 16 + l16;
    const bool n0ok = n0i < N;
    const bool n1ok = n1i < N;
    const float* __restrict__ B0 = Bm + (long long)(n0ok ? n0i : (N - 1)) * ldb + (half << 1);
    const float* __restrict__ B1 = Bm + (long long)(n1ok ? n1i : (N - 1)) * ldb + (half << 1);

    v8f acc00 = {0.f,0.f,0.f,0.f,0.f,0.f,0.f,0.f};
    v8f acc01 = acc00, acc10 = acc00, acc11 = acc00;

#pragma unroll 4
    for (int k = 0; k < K; k += 4) {
        v2f a0 = *(const v2f*)(A0 + k);
        v2f a1 = *(const v2f*)(A1 + k);
        v2f b0 = *(const v2f*)(B0 + k);
        v2f b1 = *(const v2f*)(B1 + k);
        acc00 = wmma4(a0, b0, acc00);
        acc01 = wmma4(a0, b1, acc01);
        acc10 = wmma4(a1, b0, acc10);
        acc11 = wmma4(a1, b1, acc11);
    }

    const float bv0 = (bias && n0ok) ? bias[n0i] : 0.f;
    const float bv1 = (bias && n1ok) ? bias[n1i] : 0.f;

    store_tile(acc00, m0,      half, n0i, n0ok, bv0, C, ldc, act, row_remap);
    store_tile(acc01, m0,      half, n1i, n1ok, bv1, C, ldc, act, row_remap);
    store_tile(acc10, m0 + 16, half, n0i, n0ok, bv0, C, ldc, act, row_remap);
    store_tile(acc11, m0 + 16, half, n1i, n1ok, bv1, C, ldc, act, row_remap);
}

// ---------------------------------------------------------------------------
// Prep: Wcat[2048,1024] = [Wih | Whh], bsum = bih + bhh
// ---------------------------------------------------------------------------
__global__ void build_wcat(const float* __restrict__ Wih, const float* __restrict__ Whh,
                           const float* __restrict__ bih, const float* __restrict__ bhh,
                           float* __restrict__ wcat, float* __restrict__ bsum)
{
    int idx = blockIdx.x * blockDim.x + threadIdx.x;
    if (idx >= 2048 * 1024) return;
    int n = idx >> 10, k = idx & 1023;
    wcat[idx] = (k < 512) ? Wih[n * 512 + k] : Whh[n * 512 + (k - 512)];
    if (k == 0) bsum[n] = bih[n] + bhh[n];
}

__global__ void build_fccat(const float* __restrict__ f, const float* __restrict__ l,
                            float* __restrict__ out)
{
    int idx = blockIdx.x * blockDim.x + threadIdx.x;
    if (idx >= BSZ * 1024) return;
    int b = idx >> 10, k = idx & 1023;
    out[idx] = (k < 512) ? f[b * 512 + k] : l[b * 512 + (k - 512)];
}

// ---------------------------------------------------------------------------
// im2col for conv1 (input = embed lookup, NO relu) : xcol[(b*48+l)][i*3+k]
// ---------------------------------------------------------------------------
__global__ void im2col_embed(const int* __restrict__ cap, int sidx,
                             const float* __restrict__ embed,
                             float* __restrict__ xcol)
{
    int idx = blockIdx.x * blockDim.x + threadIdx.x;
    const int total = BSZ * 48 * 1536;
    if (idx >= total) return;
    int row = idx / 1536, col = idx - row * 1536;
    int b = row / 48, l = row - b * 48;
    int i = col / 3, k = col - i * 3;
    int tok = cap[(b * SMAX + sidx) * NMAX + l + k];
    xcol[idx] = embed[tok * DSZ + i];
}

// im2col for conv2/conv3 (input = previous conv output [B*Lin, 512], relu applied)
__global__ void im2col_prev(const float* __restrict__ prev,
                            float* __restrict__ xcol, int Lin, int Lout)
{
    int idx = blockIdx.x * blockDim.x + threadIdx.x;
    const int total = BSZ * Lout * 1536;
    if (idx >= total) return;
    int row = idx / 1536, col = idx - row * 1536;
    int b = row / Lout, l = row - b * Lout;
    int i = col / 3, k = col - i * 3;
    xcol[idx] = prev[((long long)(b * Lin + l + k)) * DSZ + i];
}

// max over sequence positions: sv[b*1536 + off + o] = max_l X[(b*L+l)*512 + o]
__global__ void colmax(const float* __restrict__ X, float* __restrict__ sv,
                       int L, int off)
{
    int idx = blockIdx.x * blockDim.x + threadIdx.x;
    if (idx >= BSZ * DSZ) return;
    int b = idx >> 9, o = idx & 511;
    float m = -3.0e38f;
    for (int l = 0; l < L; ++l)
        m = fmaxf(m, X[((long long)(b * L + l)) * DSZ + o]);
    sv[b * 1536 + off + o] = m;
}

// ---------------------------------------------------------------------------
// Additive attention (per batch element): score_a = alpha . tanh(p_att + att_h)
// softmax over A=49, v = sum_a w_a * att_feats[b,a,:]
// one block (256 thr, 8 waves) per b
// ---------------------------------------------------------------------------
__global__ void attention_kernel(const float* __restrict__ p_att,
                                 const float* __restrict__ att_feats,
                                 const float* __restrict__ atth,
                                 const float* __restrict__ alpha_W,
                                 const float* __restrict__ alpha_b,
                                 float* __restrict__ vout, int vld, int voff)
{
    const int b = blockIdx.x;
    const int wave = threadIdx.x >> 5;
    const int lane = threadIdx.x & 31;
    __shared__ float score[ASZ];

    for (int a = wave; a < ASZ; a += 8) {
        const float* pa = p_att + ((long long)(b * ASZ + a)) * DSZ;
        const float* hh = atth + b * DSZ;
        float part = 0.f;
        for (int d = lane; d < DSZ; d += 32)
            part += tanhf(pa[d] + hh[d]) * alpha_W[d];
        for (int off = 16; off; off >>= 1) part += __shfl_xor(part, off, 32);
        if (lane == 0) score[a] = part + alpha_b[0];
    }
    __syncthreads();

    if (threadIdx.x < 32) {
        float m = -3.0e38f;
        for (int a = lane; a < ASZ; a += 32) m = fmaxf(m, score[a]);
        for (int off = 16; off; off >>= 1) m = fmaxf(m, __shfl_xor(m, off, 32));
        float ssum = 0.f;
        for (int a = lane; a < ASZ; a += 32) {
            float e = __expf(score[a] - m); score[a] = e; ssum += e;
        }
        for (int off = 16; off; off >>= 1) ssum += __shfl_xor(ssum, off, 32);
        float inv = 1.f / ssum;
        for (int a = lane; a < ASZ; a += 32) score[a] *= inv;
    }
    __syncthreads();

    for (int d = threadIdx.x; d < DSZ; d += 256) {
        float acc = 0.f;
        const float* af = att_feats + ((long long)b * ASZ) * DSZ + d;
        for (int a = 0; a < ASZ; ++a) acc += score[a] * af[(long long)a * DSZ];
        vout[b * vld + voff + d] = acc;
    }
}

// ---------------------------------------------------------------------------
// Seed decode state: XH[:,0:512]=relu(embed[tok(b,s,0)]), XH[:,512:]=h0, C=c0
// row = s*64+b, one block per row
// ---------------------------------------------------------------------------
__global__ void seed_decode(const float* __restrict__ h0s, const float* __restrict__ c0s,
                            const float* __restrict__ embed, const int* __restrict__ cap,
                            float* __restrict__ XH, float* __restrict__ Cst)
{
    int row = blockIdx.x;
    int s = row >> 6, b = row & 63;
    int tok = cap[(b * SMAX + s) * NMAX + 0];
    for (int col = threadIdx.x; col < DSZ; col += 256) {
        Cst[row * DSZ + col] = c0s[row * DSZ + col];
        XH[row * 1024 + 512 + col] = h0s[row * DSZ + col];
        XH[row * 1024 + col] = fmaxf(embed[tok * DSZ + col], 0.f);
    }
}

// ---------------------------------------------------------------------------
// LSTM pointwise: gates -> (c,h); h into XH[:,512:]; next token embed into XH[:,0:512]
// ---------------------------------------------------------------------------
__global__ void lstm_point(const float* __restrict__ G, float* __restrict__ Cst,
                           float* __restrict__ XH, const float* __restrict__ embed,
                           const int* __restrict__ cap, int t)
{
    int row = blockIdx.x;
    int s = row >> 6, b = row & 63;
    int tok = (t + 1 < TT) ? cap[(b * SMAX + s) * NMAX + t + 1] : 0;
    const float* g = G + (long long)row * 2048;
    for (int col = threadIdx.x; col < DSZ; col += 256) {
        float gi = g[col], gf = g[512 + col], gg = g[1024 + col], go = g[1536 + col];
        float c = sigmoidf_(gf) * Cst[row * DSZ + col] + sigmoidf_(gi) * tanhf(gg);
        float h = sigmoidf_(go) * tanhf(c);
        Cst[row * DSZ + col] = c;
        XH[row * 1024 + 512 + col] = h;
        if (t + 1 < TT)
            XH[row * 1024 + col] = fmaxf(embed[tok * DSZ + col], 0.f);
    }
}

// ===========================================================================
extern "C" void kernel_launch(void* const* d_in, const int* in_sizes, int n_in,
                              void* d_out, int out_size, void* d_ws, size_t ws_size,
                              hipStream_t stream)
{
    (void)in_sizes; (void)n_in; (void)out_size; (void)ws_size;
    const float* fc_f     = (const float*)d_in[0];
    const float* att_f    = (const float*)d_in[1];
    const float* fc_l     = (const float*)d_in[2];
    const float* att_l    = (const float*)d_in[3];
    const int*   cap      = (const int*)  d_in[4];
    const float* embed_W  = (const float*)d_in[5];
    const float* logit_W  = (const float*)d_in[6];
    const float* logit_b  = (const float*)d_in[7];
    const float* c2a_fW   = (const float*)d_in[8];
    const float* c2a_fb   = (const float*)d_in[9];
    const float* c2a_lW   = (const float*)d_in[10];
    const float* c2a_lb   = (const float*)d_in[11];
    const float* Wih      = (const float*)d_in[12];
    const float* Whh      = (const float*)d_in[13];
    const float* bih      = (const float*)d_in[14];
    const float* bhh      = (const float*)d_in[15];
    const float* conv1_W  = (const float*)d_in[16];
    const float* conv1_b  = (const float*)d_in[17];
    const float* conv2_W  = (const float*)d_in[18];
    const float* conv2_b  = (const float*)d_in[19];
    const float* conv3_W  = (const float*)d_in[20];
    const float* conv3_b  = (const float*)d_in[21];
    const float* ih_W     = (const float*)d_in[22];  // [8,512,1024]
    const float* ih_b     = (const float*)d_in[23];  // [8,512]
    const float* ic_W     = (const float*)d_in[24];
    const float* ic_b     = (const float*)d_in[25];
    const float* h2a_fW   = (const float*)d_in[26];  // [512,1536]
    const float* h2a_fb   = (const float*)d_in[27];
    const float* h2a_lW   = (const float*)d_in[28];
    const float* h2a_lb   = (const float*)d_in[29];
    const float* alpha_W  = (const float*)d_in[30];
    const float* alpha_b  = (const float*)d_in[31];
    float* out = (float*)d_out;

    // ---- workspace bump allocator (floats, 16B aligned) ----
    float* wp = (float*)d_ws;
    auto alloc = [&](size_t n) { float* p = wp; wp += ((n + 3) & ~(size_t)3); return p; };
    float* wcat  = alloc(2048 * 1024);
    float* bsum  = alloc(2048);
    float* pattf = alloc((size_t)BSZ * ASZ * DSZ);
    float* pattl = alloc((size_t)BSZ * ASZ * DSZ);
    float* xcol  = alloc((size_t)BSZ * 48 * 1536);
    float* outA  = alloc((size_t)BSZ * 48 * DSZ);
    float* outB  = alloc((size_t)BSZ * 48 * DSZ);
    float* sv    = alloc((size_t)BSZ * 1536);
    float* atth  = alloc((size_t)BSZ * DSZ);
    float* vbuf  = alloc((size_t)BSZ * 1024);
    float* fccat = alloc((size_t)BSZ * 1024);
    float* h0s   = alloc((size_t)MROWS * DSZ);
    float* c0s   = alloc((size_t)MROWS * DSZ);
    float* XH    = alloc((size_t)MROWS * 1024);
    float* Cst   = alloc((size_t)MROWS * DSZ);
    float* G     = alloc((size_t)MROWS * 2048);

    auto gemm = [&](const float* A, int lda, const float* Bm, int ldb,
                    const float* bias, float* C, long long ldc,
                    int M, int N, int K, int act, int remap) {
        int tiles = (M >> 5) * ((N + 31) >> 5);   // 32x32 supertiles, one per wave
        int blocks = (tiles + 7) / 8;
        gemm_wmma_f32<<<blocks, 256, 0, stream>>>(A, lda, Bm, ldb, bias, C, ldc,
                                                  M, N, K, act, remap);
    };

    // ---- prep ----
    build_wcat<<<(2048 * 1024 + 255) / 256, 256, 0, stream>>>(Wih, Whh, bih, bhh, wcat, bsum);
    build_fccat<<<(BSZ * 1024 + 255) / 256, 256, 0, stream>>>(fc_f, fc_l, fccat);

    // ---- p_att = att_feats @ ctx2att^T + b  ([3136,512]) ----
    gemm(att_f, DSZ, c2a_fW, DSZ, c2a_fb, pattf, DSZ, BSZ * ASZ, DSZ, DSZ, 0, 0);
    gemm(att_l, DSZ, c2a_lW, DSZ, c2a_lb, pattl, DSZ, BSZ * ASZ, DSZ, DSZ, 0, 0);

    // ---- sentence 0 init: tanh(fccat @ init_W[0]^T + b) ----
    gemm(fccat, 1024, ih_W, 1024, ih_b, h0s, DSZ, BSZ, DSZ, 1024, 2, 0);
    gemm(fccat, 1024, ic_W, 1024, ic_b, c0s, DSZ, BSZ, DSZ, 1024, 2, 0);

    // ---- sentences 1..7: encode prev sentence -> attention -> init h/c ----
    for (int s = 1; s < SMAX; ++s) {
        const int L1 = 48, L2 = 46, L3 = 44;
        // conv1
        im2col_embed<<<(BSZ * L1 * 1536 + 255) / 256, 256, 0, stream>>>(cap, s - 1, embed_W, xcol);
        gemm(xcol, 1536, conv1_W, 1536, conv1_b, outA, DSZ, BSZ * L1, DSZ, 1536, 1, 0);
        colmax<<<(BSZ * DSZ + 255) / 256, 256, 0, stream>>>(outA, sv, L1, 0);
        // conv2
        im2col_prev<<<(BSZ * L2 * 1536 + 255) / 256, 256, 0, stream>>>(outA, xcol, L1, L2);
        gemm(xcol, 1536, conv2_W, 1536, conv2_b, outB, DSZ, BSZ * L2, DSZ, 1536, 1, 0);
        colmax<<<(BSZ * DSZ + 255) / 256, 256, 0, stream>>>(outB, sv, L2, 512);
        // conv3
        im2col_prev<<<(BSZ * L3 * 1536 + 255) / 256, 256, 0, stream>>>(outB, xcol, L2, L3);
        gemm(xcol, 1536, conv3_W, 1536, conv3_b, outA, DSZ, BSZ * L3, DSZ, 1536, 1, 0);
        colmax<<<(BSZ * DSZ + 255) / 256, 256, 0, stream>>>(outA, sv, L3, 1024);

        // attention frontal / lateral
        gemm(sv, 1536, h2a_fW, 1536, h2a_fb, atth, DSZ, BSZ, DSZ, 1536, 0, 0);
        attention_kernel<<<BSZ, 256, 0, stream>>>(pattf, att_f, atth, alpha_W, alpha_b,
                                                  vbuf, 1024, 0);
        gemm(sv, 1536, h2a_lW, 1536, h2a_lb, atth, DSZ, BSZ, DSZ, 1536, 0, 0);
        attention_kernel<<<BSZ, 256, 0, stream>>>(pattl, att_l, atth, alpha_W, alpha_b,
                                                  vbuf, 1024, 512);

        // init h0/c0 for this sentence
        gemm(vbuf, 1024, ih_W + (size_t)s * DSZ * 1024, 1024, ih_b + s * DSZ,
             h0s + (size_t)s * BSZ * DSZ, DSZ, BSZ, DSZ, 1024, 2, 0);
        gemm(vbuf, 1024, ic_W + (size_t)s * DSZ * 1024, 1024, ic_b + s * DSZ,
             c0s + (size_t)s * BSZ * DSZ, DSZ, BSZ, DSZ, 1024, 2, 0);
    }

    // ---- fused decode: all 8 sentences as one M=512 batch, 49 sequential steps ----
    seed_decode<<<MROWS, 256, 0, stream>>>(h0s, c0s, embed_W, cap, XH, Cst);
    for (int t = 0; t < TT; ++t) {
        // gates: [512,1024] @ [2048,1024]^T
        gemm(XH, 1024, wcat, 1024, bsum, G, 2048, MROWS, 2048, 1024, 0, 0);
        lstm_point<<<MROWS, 256, 0, stream>>>(G, Cst, XH, embed_W, cap, t);
        // logits: [512,512] @ [2070,512]^T, scattered to out[b,s,t,:]
        gemm(XH + 512, 1024, logit_W, DSZ, logit_b,
             out + (long long)t * VSZ, (long long)TT * VSZ,
             MROWS, VSZ, DSZ, 0, 1);
    }
}